// RPN_24326694764922
// MI455X (gfx1250) — compile-verified
//
#include <hip/hip_runtime.h>
#include <hip/hip_bf16.h>

typedef __attribute__((ext_vector_type(2))) float v2f;
typedef __attribute__((ext_vector_type(8))) float v8f;

#define CIN   512
#define COUT  512
#define M_TOT 1024          // 32*32 spatial positions
#define KTOT  4608          // 512*9
#define NANCH 9
#define NBOX  9216          // 1024*9
#define NMS_THR 0.7f

// ---- workspace layout (float offsets) ----
#define OFF_WT    0                          // 4608*512   = 2359296
#define OFF_FEATS (OFF_WT + KTOT*COUT)       // 1024*512   =  524288
#define OFF_CLSP  (OFF_FEATS + M_TOT*COUT)   // 9216*2
#define OFF_SCORE (OFF_CLSP + NBOX*2)        // 9216
#define OFF_BOX   (OFF_SCORE + NBOX)         // 9216*4

// LDS pads: keep fragment (k,k+1) pairs contiguous + bank-spread rows
#define APAD 68             // As row stride (floats): 4 mod 64 banks
#define BPAD 66             // Bs row stride (floats): 2 mod 64 banks

// ============================================================
// Kernel 0: weights (co,ci,3,3) -> B matrix wT[k][co], k = j*512+ci
// ============================================================
__global__ __launch_bounds__(512)
void k_transpose_w(const float* __restrict__ w, float* __restrict__ wT) {
    const int k  = blockIdx.x;        // 0..4607
    const int j  = k >> 9;            // 0..8  (ky*3+kx)
    const int ci = k & 511;
    const int co = threadIdx.x;       // 0..511
    wT[k * COUT + co] = w[(co * CIN + ci) * 9 + j];
}

// ============================================================
// Kernel 1: 3x3 conv + bias + relu via implicit GEMM on
// V_WMMA_F32_16X16X4_F32.
// Block = 4 waves, block tile 32M x 64N; wave = 16M x 32N with two
// accumulator chains sharing one A fragment.
// LDS is m-/n-major so every fragment is ONE contiguous ds_load_b64
// (no register re-shuffle movs in the WMMA loop).
// ============================================================
__global__ __launch_bounds__(128)
void k_conv3x3(const float* __restrict__ x, const float* __restrict__ wT,
               const float* __restrict__ bias, float* __restrict__ feats) {
    __shared__ __align__(16) float As[32][APAD];   // As[m][k], k contiguous
    __shared__ __align__(16) float Bs[64][BPAD];   // Bs[n][k], k contiguous

    const int t     = threadIdx.x;
    const int lane  = t & 31;
    const int wv    = t >> 5;               // wave 0..3
    const int msub  = (wv >> 1) * 16;       // wave M sub-tile: 0 / 16
    const int nsub  = (wv & 1) * 32;        // wave N sub-tile: 0 / 32
    const int hi16  = (lane >= 16) ? 1 : 0; // upper half-wave: K+2 / M+8
    const int ml    = lane & 15;
    const int mtile = blockIdx.y;           // 0..31 (32 M rows per block)
    const int nbase = blockIdx.x * 64;      // 0..7 * 64

    const float* arow = &As[msub + ml][0];        // this lane's A row
    const float* brow0 = &Bs[nsub + ml][0];       // fragment-0 B row
    const float* brow1 = &Bs[nsub + 16 + ml][0];  // fragment-1 B row

    v8f acc0 = {};
    v8f acc1 = {};
    for (int kb = 0; kb < 72; ++kb) {       // 72 * 64 = 4608
        const int j   = kb >> 3;            // filter tap 0..8
        const int cib = (kb & 7) * 64;      // cin base within tap
        const int dy  = j / 3 - 1;
        const int dx  = j % 3 - 1;

        // stage A tile: 32m x 64k (im2col on the fly, zero-padded)
        for (int r = 0; r < 16; ++r) {
            int idx = r * 128 + t;
            int kl  = idx >> 5;
            int m   = idx & 31;             // fastest -> coalesced global read
            int mg  = mtile * 32 + m;
            int yy  = (mg >> 5) + dy;
            int xx  = (mg & 31) + dx;
            float v = 0.0f;
            if (yy >= 0 && yy < 32 && xx >= 0 && xx < 32)
                v = x[(cib + kl) * 1024 + yy * 32 + xx];
            As[m][kl] = v;
        }
        // stage B tile: 64n x 64k (coalesced from transposed weights)
        const float* wTp = wT + (j * 512 + cib) * COUT + nbase;
        for (int r = 0; r < 32; ++r) {
            int idx = r * 128 + t;
            int kl  = idx >> 6;
            int n   = idx & 63;             // fastest -> coalesced global read
            Bs[n][kl] = wTp[kl * COUT + n];
        }
        // prefetch next K-block's B tile toward this WGP while we compute
        if (kb + 1 < 72) {
            const int jn  = (kb + 1) >> 3;
            const int cn  = ((kb + 1) & 7) * 64;
            const float* pn = wT + (jn * 512 + cn) * COUT + nbase;
            __builtin_prefetch(pn + (t >> 1) * COUT + (t & 1) * 32, 0, 0);
        }
        __syncthreads();

        // 16 K-steps of 4; 3 ds_load_b64 + 2 WMMA per step
        for (int kk = 0; kk < 16; ++kk) {
            const int krow = kk * 4 + hi16 * 2;   // even -> 8B aligned
            v2f a  = *(const v2f*)(arow  + krow);
            v2f b0 = *(const v2f*)(brow0 + krow);
            v2f b1 = *(const v2f*)(brow1 + krow);
            acc0 = __builtin_amdgcn_wmma_f32_16x16x4_f32(
                       false, a, false, b0, (short)0, acc0, false, false);
            acc1 = __builtin_amdgcn_wmma_f32_16x16x4_f32(
                       false, a, false, b1, (short)0, acc1, false, false);
        }
        __syncthreads();
    }

    // C/D layout: VGPR v -> M = v + 8*hi16, N = ml
    const int   ng0 = nbase + nsub + ml;
    const float bb0 = bias[ng0];
    const float bb1 = bias[ng0 + 16];
    for (int v = 0; v < 8; ++v) {
        int   mg = mtile * 32 + msub + v + hi16 * 8;
        float f0 = acc0[v] + bb0;
        float f1 = acc1[v] + bb1;
        feats[mg * COUT + ng0]      = f0 > 0.0f ? f0 : 0.0f;
        feats[mg * COUT + ng0 + 16] = f1 > 0.0f ? f1 : 0.0f;
    }
}

// ============================================================
// Kernel 2: fused 1x1 cls/reg convs + softmax + anchors + boxes
// One block per spatial position m; 54 dot products of K=512.
// ============================================================
__global__ __launch_bounds__(64)
void k_heads(const float* __restrict__ feats,
             const float* __restrict__ w_cls, const float* __restrict__ b_cls,
             const float* __restrict__ w_reg, const float* __restrict__ b_reg,
             const float* __restrict__ anchors_wh,
             float* __restrict__ cls_prob, float* __restrict__ scores,
             float* __restrict__ boxes) {
    __shared__ float logits[64];
    const int m = blockIdx.x;
    const int n = threadIdx.x;
    const float* f = feats + m * COUT;

    if (n < 54) {
        const float* wp = (n < 18) ? (w_cls + n * COUT) : (w_reg + (n - 18) * COUT);
        float s = (n < 18) ? b_cls[n] : b_reg[n - 18];
        for (int k = 0; k < COUT; ++k) s = fmaf(f[k], wp[k], s);
        logits[n] = s;
    }
    __syncthreads();

    if (n < NANCH) {
        const int idx = m * NANCH + n;
        // softmax over (2n, 2n+1); score = P(class 0)
        float c0 = logits[2 * n], c1 = logits[2 * n + 1];
        float mx = fmaxf(c0, c1);
        float e0 = expf(c0 - mx), e1 = expf(c1 - mx);
        float inv = 1.0f / (e0 + e1);
        cls_prob[idx * 2 + 0] = e0 * inv;
        cls_prob[idx * 2 + 1] = e1 * inv;
        scores[idx] = e0 * inv;
        // deparameterize
        float r0 = logits[18 + 4 * n + 0];
        float r1 = logits[18 + 4 * n + 1];
        float r2 = logits[18 + 4 * n + 2];
        float r3 = logits[18 + 4 * n + 3];
        const float inv32 = 1.0f / 32.0f;
        float wa = anchors_wh[2 * n + 0] * inv32;
        float ha = anchors_wh[2 * n + 1] * inv32;
        float xa = (float)(m >> 5) * inv32;   // first spatial dim
        float ya = (float)(m & 31) * inv32;   // second spatial dim
        float xc = r0 * wa + xa;
        float yc = r1 * ha + ya;
        float bw = expf(r2) * wa * 0.5f;
        float bh = expf(r3) * ha * 0.5f;
        boxes[idx * 4 + 0] = xc - bw;
        boxes[idx * 4 + 1] = yc - bh;
        boxes[idx * 4 + 2] = xc + bw;
        boxes[idx * 4 + 3] = yc + bh;
    }
}

// ============================================================
// Kernel 3: greedy top-5 NMS (equivalent to reference sort+scan),
// single block, suppression bitmap in LDS.
// ============================================================
__global__ __launch_bounds__(256)
void k_nms(const float* __restrict__ scores, const float* __restrict__ boxes,
           const float* __restrict__ cls_prob, float* __restrict__ out) {
    __shared__ unsigned char sup[NBOX];
    __shared__ float sV[256];
    __shared__ int   sI[256];
    __shared__ int   sBest;
    const int t = threadIdx.x;

    for (int j = t; j < NBOX; j += 256) sup[j] = 0;
    __syncthreads();

    for (int pick = 0; pick < 5; ++pick) {
        // argmax over unsuppressed (tie -> lowest index, matches stable sort)
        float bv = -1.0f; int bi = NBOX;
        for (int j = t; j < NBOX; j += 256) {
            if (!sup[j]) {
                float s = scores[j];
                if (s > bv || (s == bv && j < bi)) { bv = s; bi = j; }
            }
        }
        sV[t] = bv; sI[t] = bi;
        __syncthreads();
        for (int st = 128; st > 0; st >>= 1) {
            if (t < st) {
                float v2 = sV[t + st]; int i2 = sI[t + st];
                if (v2 > sV[t] || (v2 == sV[t] && i2 < sI[t])) { sV[t] = v2; sI[t] = i2; }
            }
            __syncthreads();
        }
        if (t == 0) {
            int best = sI[0];
            sBest = best;
            sup[best] = 1;
            out[2 * pick + 0]      = cls_prob[best * 2 + 0];
            out[2 * pick + 1]      = cls_prob[best * 2 + 1];
            out[10 + 4 * pick + 0] = boxes[best * 4 + 0];
            out[10 + 4 * pick + 1] = boxes[best * 4 + 1];
            out[10 + 4 * pick + 2] = boxes[best * 4 + 2];
            out[10 + 4 * pick + 3] = boxes[best * 4 + 3];
        }
        __syncthreads();
        const int best = sBest;
        const float ax1 = boxes[best * 4 + 0], ay1 = boxes[best * 4 + 1];
        const float ax2 = boxes[best * 4 + 2], ay2 = boxes[best * 4 + 3];
        const float areaA = (ax2 - ax1) * (ay2 - ay1);
        for (int j = t; j < NBOX; j += 256) {
            if (!sup[j]) {
                float bx1 = boxes[j * 4 + 0], by1 = boxes[j * 4 + 1];
                float bx2 = boxes[j * 4 + 2], by2 = boxes[j * 4 + 3];
                float iw = fminf(ax2, bx2) - fmaxf(ax1, bx1);
                float ih = fminf(ay2, by2) - fmaxf(ay1, by1);
                iw = iw > 0.0f ? iw : 0.0f;
                ih = ih > 0.0f ? ih : 0.0f;
                float inter = iw * ih;
                float areaB = (bx2 - bx1) * (by2 - by1);
                float iou = inter / (areaA + areaB - inter);
                if (iou > NMS_THR) sup[j] = 1;
            }
        }
        __syncthreads();
    }
}

// ============================================================
extern "C" void kernel_launch(void* const* d_in, const int* in_sizes, int n_in,
                              void* d_out, int out_size, void* d_ws, size_t ws_size,
                              hipStream_t stream) {
    (void)in_sizes; (void)n_in; (void)out_size; (void)ws_size;
    const float* x          = (const float*)d_in[0];
    const float* w_win      = (const float*)d_in[1];
    const float* b_win      = (const float*)d_in[2];
    const float* w_cls      = (const float*)d_in[3];
    const float* b_cls      = (const float*)d_in[4];
    const float* w_reg      = (const float*)d_in[5];
    const float* b_reg      = (const float*)d_in[6];
    const float* anchors_wh = (const float*)d_in[7];

    float* ws       = (float*)d_ws;
    float* wT       = ws + OFF_WT;
    float* feats    = ws + OFF_FEATS;
    float* cls_prob = ws + OFF_CLSP;
    float* scores   = ws + OFF_SCORE;
    float* boxes    = ws + OFF_BOX;
    float* out      = (float*)d_out;

    k_transpose_w<<<dim3(KTOT), 512, 0, stream>>>(w_win, wT);
    k_conv3x3<<<dim3(8, 32), 128, 0, stream>>>(x, wT, b_win, feats);
    k_heads<<<dim3(M_TOT), 64, 0, stream>>>(feats, w_cls, b_cls, w_reg, b_reg,
                                            anchors_wh, cls_prob, scores, boxes);
    k_nms<<<dim3(1), 256, 0, stream>>>(scores, boxes, cls_prob, out);
}